// GM_GCN_51780125721472
// MI455X (gfx1250) — compile-verified
//
#include <hip/hip_runtime.h>
#include <hip/hip_bf16.h>

typedef __attribute__((ext_vector_type(2))) float v2f;
typedef __attribute__((ext_vector_type(8))) float v8f;

#define D_FEAT 128
#define ROW_TILES 5              // 5 x 16 = 80 rows per block; 50000 = 625 * 80

__device__ __forceinline__ void atomic_add_f32(float* p, float v) {
    unsafeAtomicAdd(p, v);       // guaranteed hardware global_atomic_add_f32
}

// ---------------- degree / norm ----------------

__global__ void init_deg_kernel(float* deg, int N) {
    int i = blockIdx.x * blockDim.x + threadIdx.x;
    if (i < N) deg[i] = 1.0f;  // self-loop contributes 1
}

__global__ void add_deg_kernel(const int* __restrict__ col, float* deg, int E) {
    int e = blockIdx.x * blockDim.x + threadIdx.x;
    if (e < E) atomic_add_f32(&deg[col[e]], 1.0f);
}

__global__ void rsqrt_kernel(float* deg, int N) {
    int i = blockIdx.x * blockDim.x + threadIdx.x;
    if (i < N) deg[i] = rsqrtf(deg[i]);  // deg >= 1 always (self-loop)
}

__global__ void norm_kernel(const int* __restrict__ row, const int* __restrict__ col,
                            const float* __restrict__ dinv, float* __restrict__ norm, int E) {
    int e = blockIdx.x * blockDim.x + threadIdx.x;
    if (e < E) norm[e] = dinv[row[e]] * dinv[col[e]];
}

// ---------------- dense GEMM via V_WMMA_F32_16X16X4_F32 ----------------
// Y[N x 128] = (RELU_IN ? relu(X) : X)[N x 128] @ W[128 x 128]
// Block = 256 threads = 8 waves. Each wave computes 5 row-tiles (80x16 output):
// B is loaded once per K-step and reused by 5 independent WMMAs.
// Grid.x = N/80. EXEC is all-ones at every WMMA.

template <bool RELU_IN>
__global__ __launch_bounds__(256) void gemm128_kernel(const float* __restrict__ X,
                                                      const float* __restrict__ W,
                                                      float* __restrict__ Y) {
    const int wave = threadIdx.x >> 5;
    const int lane = threadIdx.x & 31;
    const int row0 = blockIdx.x * (16 * ROW_TILES);
    const int col0 = wave * 16;
    const int m  = lane & 15;   // M (A) or N (B/C) index within tile
    const int hi = lane >> 4;   // lane half: selects K sub-pair

    const float* xbase = X + (size_t)(row0 + m) * D_FEAT;

    v8f c[ROW_TILES];
    #pragma unroll
    for (int t = 0; t < ROW_TILES; ++t) c[t] = (v8f){};

    #pragma unroll 2
    for (int k0 = 0; k0 < D_FEAT; k0 += 4) {
        const int ka = k0 + 2 * hi;
        // B 4x16: row K striped across lanes within a VGPR, mirrored halves
        v2f b;
        b.x = W[(size_t)(ka + 0) * D_FEAT + col0 + m];
        b.y = W[(size_t)(ka + 1) * D_FEAT + col0 + m];
        // A 16x4 per tile: lanes0-15 hold K=k0,k0+1 ; lanes16-31 K=k0+2,k0+3
        #pragma unroll
        for (int t = 0; t < ROW_TILES; ++t) {
            v2f a = *(const v2f*)(xbase + (size_t)t * 16 * D_FEAT + ka);
            if (RELU_IN) { a.x = fmaxf(a.x, 0.0f); a.y = fmaxf(a.y, 0.0f); }
            c[t] = __builtin_amdgcn_wmma_f32_16x16x4_f32(false, a, false, b, (short)0,
                                                         c[t], false, false);
        }
    }
    // C/D 16x16 f32: VGPR i holds M = i + 8*hi, N = m
    #pragma unroll
    for (int t = 0; t < ROW_TILES; ++t) {
        #pragma unroll
        for (int i = 0; i < 8; ++i) {
            Y[(size_t)(row0 + t * 16 + i + 8 * hi) * D_FEAT + col0 + m] = c[t][i];
        }
    }
}

// Final classifier: Y[N x 40] = relu(X) @ Wout[128 x 40] + bout
// 3 waves/block covering padded columns 0..47; B-loads masked arithmetically
// (clamped address * 0/1 mask) so EXEC stays all-ones at the WMMAs.
__global__ __launch_bounds__(96) void gemm_out_kernel(const float* __restrict__ X,
                                                      const float* __restrict__ W,
                                                      const float* __restrict__ bias,
                                                      float* __restrict__ Y, int C) {
    const int wave = threadIdx.x >> 5;  // 0..2
    const int lane = threadIdx.x & 31;
    const int row0 = blockIdx.x * (16 * ROW_TILES);
    const int m  = lane & 15;
    const int hi = lane >> 4;
    const int col  = wave * 16 + m;           // 0..47
    const int colc = (col < C) ? col : 0;     // clamped address
    const float mask = (col < C) ? 1.0f : 0.0f;

    const float* xbase = X + (size_t)(row0 + m) * D_FEAT;

    v8f c[ROW_TILES];
    #pragma unroll
    for (int t = 0; t < ROW_TILES; ++t) c[t] = (v8f){};

    #pragma unroll 2
    for (int k0 = 0; k0 < D_FEAT; k0 += 4) {
        const int ka = k0 + 2 * hi;
        v2f b;
        b.x = mask * W[(size_t)(ka + 0) * C + colc];
        b.y = mask * W[(size_t)(ka + 1) * C + colc];
        #pragma unroll
        for (int t = 0; t < ROW_TILES; ++t) {
            v2f a = *(const v2f*)(xbase + (size_t)t * 16 * D_FEAT + ka);
            a.x = fmaxf(a.x, 0.0f);
            a.y = fmaxf(a.y, 0.0f);
            c[t] = __builtin_amdgcn_wmma_f32_16x16x4_f32(false, a, false, b, (short)0,
                                                         c[t], false, false);
        }
    }
    if (col < C) {
        const float bb = bias[col];
        #pragma unroll
        for (int t = 0; t < ROW_TILES; ++t) {
            #pragma unroll
            for (int i = 0; i < 8; ++i) {
                Y[(size_t)(row0 + t * 16 + i + 8 * hi) * C + col] = c[t][i] + bb;
            }
        }
    }
}

// ---------------- aggregation ----------------

// out[n][d] = bias[d] + dinv[n]^2 * tmp[n][d]   (self-loop term, pre-atomics)
__global__ void init_out_kernel(const float* __restrict__ tmp, const float* __restrict__ dinv,
                                const float* __restrict__ bias, float* __restrict__ out, int N) {
    int idx = blockIdx.x * blockDim.x + threadIdx.x;  // over N*32 float4 slots
    if (idx >= N * 32) return;
    int n = idx >> 5;
    int q = idx & 31;
    float di = dinv[n];
    float s = di * di;
    float4 t = ((const float4*)(tmp + (size_t)n * D_FEAT))[q];
    float4 bb = ((const float4*)bias)[q];
    float4 o;
    o.x = bb.x + s * t.x;
    o.y = bb.y + s * t.y;
    o.z = bb.z + s * t.z;
    o.w = bb.w + s * t.w;
    ((float4*)(out + (size_t)n * D_FEAT))[q] = o;
}

// One wave per edge: lane l handles features 4l..4l+3.
// float4 gather of tmp[row], 4x global_atomic_add_f32 into out[col].
__global__ __launch_bounds__(256) void edge_agg_kernel(const int* __restrict__ row,
                                                       const int* __restrict__ col,
                                                       const float* __restrict__ norm,
                                                       const float* __restrict__ tmp,
                                                       float* __restrict__ out, int E) {
    int e = (int)((blockIdx.x * (size_t)blockDim.x + threadIdx.x) >> 5);
    int lane = threadIdx.x & 31;
    if (e >= E) return;
    int r = row[e];
    int c = col[e];
    float w = norm[e];
    float4 v = ((const float4*)(tmp + (size_t)r * D_FEAT))[lane];
    float* dst = out + (size_t)c * D_FEAT + lane * 4;
    atomic_add_f32(dst + 0, w * v.x);
    atomic_add_f32(dst + 1, w * v.y);
    atomic_add_f32(dst + 2, w * v.z);
    atomic_add_f32(dst + 3, w * v.w);
}

// ---------------- driver ----------------

static inline size_t align256(size_t x) { return (x + 255) & ~(size_t)255; }

extern "C" void kernel_launch(void* const* d_in, const int* in_sizes, int n_in,
                              void* d_out, int out_size, void* d_ws, size_t ws_size,
                              hipStream_t stream) {
    const float* x    = (const float*)d_in[0];
    const int*   ei   = (const int*)d_in[1];
    const float* W1   = (const float*)d_in[2];
    const float* b1   = (const float*)d_in[3];
    const float* W2   = (const float*)d_in[4];
    const float* b2   = (const float*)d_in[5];
    const float* W3   = (const float*)d_in[6];
    const float* b3   = (const float*)d_in[7];
    const float* Wout = (const float*)d_in[8];
    const float* bout = (const float*)d_in[9];
    float* out = (float*)d_out;

    const int N = in_sizes[0] / D_FEAT;   // 50000
    const int E = in_sizes[1] / 2;        // 800000
    const int C = 40;
    const int* erow = ei;       // edge_index[0]
    const int* ecol = ei + E;   // edge_index[1]

    // workspace layout
    char* ws = (char*)d_ws;
    size_t off = 0;
    float* dinv = (float*)(ws + off); off = align256(off + (size_t)N * 4);
    float* norm = (float*)(ws + off); off = align256(off + (size_t)E * 4);
    float* bufA = (float*)(ws + off); off = align256(off + (size_t)N * D_FEAT * 4);
    float* bufB = (float*)(ws + off); off = align256(off + (size_t)N * D_FEAT * 4);
    (void)ws_size;

    const int TB = 256;
    const int gN   = (N + TB - 1) / TB;
    const int gE   = (E + TB - 1) / TB;
    const int gRow = N / (16 * ROW_TILES);      // 625 (N % 80 == 0)
    const int gAgg = (E * 32 + TB - 1) / TB;    // one wave per edge
    const int gEl  = (N * 32 + TB - 1) / TB;    // one thread per float4

    // ---- normalization coefficients ----
    init_deg_kernel<<<gN, TB, 0, stream>>>(dinv, N);
    add_deg_kernel<<<gE, TB, 0, stream>>>(ecol, dinv, E);
    rsqrt_kernel<<<gN, TB, 0, stream>>>(dinv, N);
    norm_kernel<<<gE, TB, 0, stream>>>(erow, ecol, dinv, norm, E);

    // ---- layer 1: x -> bufA (gemm) -> bufB (aggregate) ----
    gemm128_kernel<false><<<gRow, TB, 0, stream>>>(x, W1, bufA);
    init_out_kernel<<<gEl, TB, 0, stream>>>(bufA, dinv, b1, bufB, N);
    edge_agg_kernel<<<gAgg, TB, 0, stream>>>(erow, ecol, norm, bufA, bufB, E);

    // ---- layer 2: relu(bufB) -> bufA (gemm) -> bufB (aggregate) ----
    gemm128_kernel<true><<<gRow, TB, 0, stream>>>(bufB, W2, bufA);
    init_out_kernel<<<gEl, TB, 0, stream>>>(bufA, dinv, b2, bufB, N);
    edge_agg_kernel<<<gAgg, TB, 0, stream>>>(erow, ecol, norm, bufA, bufB, E);

    // ---- layer 3: relu(bufB) -> bufA (gemm) -> bufB (aggregate) ----
    gemm128_kernel<true><<<gRow, TB, 0, stream>>>(bufB, W3, bufA);
    init_out_kernel<<<gEl, TB, 0, stream>>>(bufA, dinv, b3, bufB, N);
    edge_agg_kernel<<<gAgg, TB, 0, stream>>>(erow, ecol, norm, bufA, bufB, E);

    // ---- classifier: relu(bufB) @ Wout + bout -> d_out ----
    gemm_out_kernel<<<gRow, 96, 0, stream>>>(bufB, Wout, bout, out, C);
    (void)out_size; (void)n_in;
}